// BPMLLLoss_24197845745967
// MI455X (gfx1250) — compile-verified
//
#include <hip/hip_runtime.h>
#include <hip/hip_bf16.h>

typedef __attribute__((ext_vector_type(2))) float v2f;
typedef __attribute__((ext_vector_type(8))) float v8f;
typedef __attribute__((ext_vector_type(2))) int   v2i;

#define B_ROWS 16384
#define L_COLS 1024

// Each wave: 16 rows x 512 cols, reduced with V_WMMA_F32_16X16X4_F32 against a
// ones B-matrix (D[m][n] = sum_k A[m][k], accumulated over the K loop).
// Block = 8 waves = 4 row-groups x 2 K-halves -> 64 rows per block.
__global__ __launch_bounds__(256) void bpmll_rows_kernel(
    const float* __restrict__ c, const int* __restrict__ y,
    float* __restrict__ losses) {
  __shared__ float lpos[8][16];
  __shared__ float lneg[8][16];
  __shared__ float lcnt[8][16];

  const int tid  = threadIdx.x;
  const int lane = tid & 31;
  const int wv   = tid >> 5;
  const int rg   = wv & 3;          // row-group within block
  const int kh   = wv >> 2;         // K half (0 or 1)

  const int m  = lane & 15;         // A-matrix row owned by this lane
  const int kb = (lane >> 4) << 1;  // A-matrix K offset: lanes 16-31 hold K=2,3

  const int row = blockIdx.x * 64 + rg * 16 + m;
  const float* crow = c + (size_t)row * L_COLS + kh * 512 + kb;
  const int*   yrow = y + (size_t)row * L_COLS + kh * 512 + kb;

  v8f cpos = {};
  v8f cneg = {};
  v2f ones;
  ones.x = 1.0f;
  ones.y = 1.0f;
  int icnt = 0;

#pragma unroll 8
  for (int k = 0; k < 512; k += 4) {
    v2f cc = *(const v2f*)(crow + k);   // 8B load, row m, cols k+kb, k+kb+1
    v2i yy = *(const v2i*)(yrow + k);
    // one exp per element: exp(-c) feeds pos (y==1), exp(+c) feeds neg (y==0)
    float e0 = __expf(yy.x ? -cc.x : cc.x);
    float e1 = __expf(yy.y ? -cc.y : cc.y);
    v2f ap, an;
    ap.x = yy.x ? e0 : 0.0f;            // y * exp(-c)
    ap.y = yy.y ? e1 : 0.0f;
    an.x = e0 - ap.x;                   // (1-y) * exp(c), disjoint with ap
    an.y = e1 - ap.y;
    icnt += yy.x + yy.y;                // integer accumulate, convert once later
    // D = A x ones + C  ->  row sums accumulate on the matrix pipe
    cpos = __builtin_amdgcn_wmma_f32_16x16x4_f32(
        false, ap, false, ones, (short)0, cpos, false, false);
    cneg = __builtin_amdgcn_wmma_f32_16x16x4_f32(
        false, an, false, ones, (short)0, cneg, false, false);
  }

  // Row m's label count lives in lanes m and m+16 (disjoint K columns).
  float cnt      = (float)icnt;
  float cnt_half = cnt + __shfl_xor(cnt, 16, 32);
  if (lane < 16) lcnt[wv][lane] = cnt_half;

  // D fragment: VGPR j holds rows j (lanes 0-15) / j+8 (lanes 16-31),
  // identical across N. One lane per half publishes its 8 row sums.
  if ((lane & 15) == 0) {
    const int half = lane >> 4;
#pragma unroll
    for (int j = 0; j < 8; ++j) {
      lpos[wv][half * 8 + j] = cpos[j];
      lneg[wv][half * 8 + j] = cneg[j];
    }
  }
  __syncthreads();

  // Merge the two K-halves (waves wv and wv+4) and finish the row.
  if (wv < 4 && lane < 16) {
    float p   = lpos[wv][lane] + lpos[wv + 4][lane];
    float n   = lneg[wv][lane] + lneg[wv + 4][lane];
    float yn  = lcnt[wv][lane] + lcnt[wv + 4][lane];
    float ybn = (float)L_COLS - yn;
    losses[blockIdx.x * 64 + wv * 16 + lane] = (p * n) / (yn * ybn);
  }
}

// Deterministic final mean over the 16384 per-row losses.
__global__ __launch_bounds__(256) void bpmll_mean_kernel(
    const float* __restrict__ v, float* __restrict__ out) {
  __shared__ float s[256];
  float acc = 0.0f;
  for (int i = threadIdx.x; i < B_ROWS; i += 256) acc += v[i];
  s[threadIdx.x] = acc;
  __syncthreads();
  for (int off = 128; off > 0; off >>= 1) {
    if (threadIdx.x < off) s[threadIdx.x] += s[threadIdx.x + off];
    __syncthreads();
  }
  if (threadIdx.x == 0) out[0] = s[0] * (1.0f / (float)B_ROWS);
}

extern "C" void kernel_launch(void* const* d_in, const int* in_sizes, int n_in,
                              void* d_out, int out_size, void* d_ws, size_t ws_size,
                              hipStream_t stream) {
  const float* c = (const float*)d_in[0];
  const int*   y = (const int*)d_in[1];
  float* losses  = (float*)d_ws;   // 16384 floats = 64 KB scratch
  float* out     = (float*)d_out;

  bpmll_rows_kernel<<<B_ROWS / 64, 256, 0, stream>>>(c, y, losses);
  bpmll_mean_kernel<<<1, 256, 0, stream>>>(losses, out);
}